// UserLinearUpscaler_70411693850629
// MI455X (gfx1250) — compile-verified
//
#include <hip/hip_runtime.h>
#include <stdint.h>

#define VOCAB 100000
#define EMBED 64
#define NB    1024
#define NS    50
#define NK    20
#define NPOS  (NB * NS)          // 51200 output positions
#define POS_PER_BLOCK 16
#define GATHER_THREADS 256       // 8 wave32s

// ---------------------------------------------------------------------------
// Main gather-reduce. One 16-lane group per position; lane owns a float4
// slice of the 64-wide embedding. Block indices staged to LDS via the CDNA5
// async global->LDS path (b128, ASYNCcnt), then K=20 coalesced b128 gathers
// from the L2-resident WT, accumulated in VGPRs.
// ---------------------------------------------------------------------------
__global__ __launch_bounds__(GATHER_THREADS)
void gather_sum(const int* __restrict__ ids, const float* __restrict__ WT,
                const float* __restrict__ bias, float* __restrict__ out) {
    alignas(16) __shared__ int s_ids[POS_PER_BLOCK * NK];  // 320 ints = 1280 B

    const int tid       = threadIdx.x;
    const int blockPos0 = blockIdx.x * POS_PER_BLOCK;

    // ---- CDNA5 async global->LDS staging of this block's 320 indices ----
    // 80 lanes x b128 (16 B each); both addresses 16 B aligned.
    {
        const uint32_t lds_base = (uint32_t)(uintptr_t)(&s_ids[0]);
        const int* gbase = ids + (size_t)blockPos0 * NK;
        if (tid < (POS_PER_BLOCK * NK) / 4) {
            const uint32_t lds_addr = lds_base + (uint32_t)(tid * 16);
            const uint32_t goff     = (uint32_t)(tid * 16);
            asm volatile("global_load_async_to_lds_b128 %0, %1, %2"
                         :: "v"(lds_addr), "v"(goff), "s"(gbase)
                         : "memory");
        }
        asm volatile("s_wait_asynccnt 0x0" ::: "memory");
    }
    __syncthreads();

    const int g   = tid >> 4;   // position group within block (0..15)
    const int l   = tid & 15;   // lane within group -> float4 slice of E
    const int pos = blockPos0 + g;

    // Pull this group's 20 ids as 5 x ds_load_b128 (80 B, 16 B aligned).
    int idbuf[NK];
    {
        const int4* sp = reinterpret_cast<const int4*>(&s_ids[g * NK]);
        #pragma unroll
        for (int c = 0; c < NK / 4; ++c) {
            const int4 q = sp[c];
            idbuf[4 * c + 0] = q.x; idbuf[4 * c + 1] = q.y;
            idbuf[4 * c + 2] = q.z; idbuf[4 * c + 3] = q.w;
        }
    }

    float4 acc = make_float4(0.f, 0.f, 0.f, 0.f);
    #pragma unroll
    for (int k = 0; k < NK; ++k) {
        const float4 v = *reinterpret_cast<const float4*>(
            WT + (size_t)idbuf[k] * EMBED + l * 4);
        acc.x += v.x; acc.y += v.y; acc.z += v.z; acc.w += v.w;
    }

    const float4 bv = *reinterpret_cast<const float4*>(bias + l * 4);
    const float4 r  = make_float4(acc.x + bv.x, acc.y + bv.y,
                                  acc.z + bv.z, acc.w + bv.w);
    *reinterpret_cast<float4*>(out + (size_t)pos * EMBED + l * 4) = r;
}

// ---------------------------------------------------------------------------
// Pre-pass: transpose W [E=64, V] (row-major) -> WT [V, E=64] so each
// gathered embedding row is 256 contiguous, 16 B-aligned bytes.
// Main path fills the LDS tile directly with async global->LDS b32 copies
// (no VGPR round-trip); the single ragged tail block uses a bounds-checked
// scalar path with zero fill. 65-float row pad -> conflict-free column reads.
// ---------------------------------------------------------------------------
__global__ __launch_bounds__(256)
void wt_transpose(const float* __restrict__ W, float* __restrict__ WT) {
    __shared__ float tile[EMBED][EMBED + 1];
    const int v0 = blockIdx.x * EMBED;
    const int tx = threadIdx.x;  // 0..63
    const int ty = threadIdx.y;  // 0..3

    if (v0 + EMBED <= VOCAB) {
        // Fast path: async global->LDS tile fill (ASYNCcnt).
        const uint32_t lds_base = (uint32_t)(uintptr_t)(&tile[0][0]);
        const float* gbase = W + v0;   // element (e, tx) at offset (e*V + tx)*4
        #pragma unroll
        for (int e = ty; e < EMBED; e += 4) {
            const uint32_t lds_addr =
                lds_base + (uint32_t)((e * (EMBED + 1) + tx) * 4);
            const uint32_t goff = (uint32_t)(e * VOCAB + tx) * 4u;  // < 2^25
            asm volatile("global_load_async_to_lds_b32 %0, %1, %2"
                         :: "v"(lds_addr), "v"(goff), "s"(gbase)
                         : "memory");
        }
        asm volatile("s_wait_asynccnt 0x0" ::: "memory");
    } else {
        #pragma unroll
        for (int e = ty; e < EMBED; e += 4) {
            const int v = v0 + tx;
            tile[e][tx] = (v < VOCAB) ? W[(size_t)e * VOCAB + v] : 0.0f;
        }
    }
    __syncthreads();

    #pragma unroll
    for (int r = ty; r < EMBED; r += 4) {
        const int v = v0 + r;
        if (v < VOCAB) WT[(size_t)v * EMBED + tx] = tile[tx][r];
    }
}

// ---------------------------------------------------------------------------
// Fallback (workspace too small for WT): direct strided gather from W.
// Correct but uncoalesced; should never be needed (WT = 25.6 MB).
// ---------------------------------------------------------------------------
__global__ __launch_bounds__(256)
void gather_direct(const int* __restrict__ ids, const float* __restrict__ W,
                   const float* __restrict__ bias, float* __restrict__ out) {
    const int e   = blockIdx.y;
    const int pos = blockIdx.x * blockDim.x + threadIdx.x;
    if (pos >= NPOS) return;
    const int* ip = ids + (size_t)pos * NK;
    float acc = 0.f;
    #pragma unroll
    for (int k = 0; k < NK; ++k)
        acc += W[(size_t)e * VOCAB + ip[k]];
    out[(size_t)pos * EMBED + e] = acc + bias[e];
}

extern "C" void kernel_launch(void* const* d_in, const int* in_sizes, int n_in,
                              void* d_out, int out_size, void* d_ws, size_t ws_size,
                              hipStream_t stream) {
    const int*   ids = (const int*)d_in[0];    // [B, S, K] int32
    const float* W   = (const float*)d_in[1];  // [E, V] f32
    const float* b   = (const float*)d_in[2];  // [E] f32
    float*       out = (float*)d_out;          // [B, S, E] f32

    const size_t wt_bytes = (size_t)VOCAB * EMBED * sizeof(float);  // 25.6 MB
    if (d_ws != nullptr && ws_size >= wt_bytes) {
        float* WT = (float*)d_ws;
        dim3 tb(64, 4);
        const int nblk = (VOCAB + EMBED - 1) / EMBED;  // 1563
        wt_transpose<<<nblk, tb, 0, stream>>>(W, WT);
        gather_sum<<<NPOS / POS_PER_BLOCK, GATHER_THREADS, 0, stream>>>(ids, WT, b, out);
    } else {
        dim3 grid((NPOS + 255) / 256, EMBED);
        gather_direct<<<grid, 256, 0, stream>>>(ids, W, b, out);
    }
}